// PrincipalAngleLoss_34428457844905
// MI455X (gfx1250) — compile-verified
//
#include <hip/hip_runtime.h>
#include <hip/hip_bf16.h>

typedef __attribute__((ext_vector_type(2))) float v2f;
typedef __attribute__((ext_vector_type(8))) float v8f;

#define D_DIM 2048
#define ROWS_PER_BLOCK 16
#define WAVES_PER_BLOCK 8
#define COLS_PER_WAVE (D_DIM / WAVES_PER_BLOCK) /* 256 */
#define CHUNK 8                                  /* columns per iteration */
#define ITERS (COLS_PER_WAVE / CHUNK)            /* 32 */

// Kernel 1: per-block (16 rows) computation of sum(theta) partials.
// Each wave handles a 256-column slice of the block's 16 rows; lanes are laid
// out to match the WMMA f32 16x16x4 A-matrix layout (lane = M + 16*Kquad), so
// the final per-row reduction is done by v_wmma_f32_16x16x4_f32 with B = ones.
__global__ __launch_bounds__(256) void pal_rowdot_kernel(
    const float* __restrict__ recon, const float* __restrict__ x,
    float* __restrict__ blockPartial)
{
    __shared__ float s_rr[WAVES_PER_BLOCK * 16];
    __shared__ float s_xx[WAVES_PER_BLOCK * 16];
    __shared__ float s_rx[WAVES_PER_BLOCK * 16];
    __shared__ float s_theta[16];

    const int tid  = threadIdx.x;
    const int wave = tid >> 5;
    const int lane = tid & 31;
    const int m    = lane & 15;   // row within the 16-row group (A-matrix M)
    const int kq   = lane >> 4;   // K quadrant: 0 -> K{0,1}, 1 -> K{2,3}

    const long rowOff = ((long)blockIdx.x * ROWS_PER_BLOCK + m) * (long)D_DIM
                      + (long)wave * COLS_PER_WAVE + (long)kq * 4;

    const float4* __restrict__ rp = (const float4*)(recon + rowOff);
    const float4* __restrict__ xp = (const float4*)(x + rowOff);

    float4 arr = make_float4(0.f, 0.f, 0.f, 0.f);
    float4 axx = make_float4(0.f, 0.f, 0.f, 0.f);
    float4 arx = make_float4(0.f, 0.f, 0.f, 0.f);

#pragma unroll 4
    for (int i = 0; i < ITERS; ++i) {
        float4 r  = rp[2 * i];   // b128 load: 4 floats of row m
        float4 xv = xp[2 * i];
        arr.x = fmaf(r.x, r.x, arr.x);
        arr.y = fmaf(r.y, r.y, arr.y);
        arr.z = fmaf(r.z, r.z, arr.z);
        arr.w = fmaf(r.w, r.w, arr.w);
        axx.x = fmaf(xv.x, xv.x, axx.x);
        axx.y = fmaf(xv.y, xv.y, axx.y);
        axx.z = fmaf(xv.z, xv.z, axx.z);
        axx.w = fmaf(xv.w, xv.w, axx.w);
        arx.x = fmaf(r.x, xv.x, arx.x);
        arx.y = fmaf(r.y, xv.y, arx.y);
        arx.z = fmaf(r.z, xv.z, arx.z);
        arx.w = fmaf(r.w, xv.w, arx.w);
    }

    // Fold 4 per-lane slots to the 2 A-matrix VGPR slots (K = 2*kq, 2*kq+1).
    v2f a_rr; a_rr.x = arr.x + arr.z; a_rr.y = arr.y + arr.w;
    v2f a_xx; a_xx.x = axx.x + axx.z; a_xx.y = axx.y + axx.w;
    v2f a_rx; a_rx.x = arx.x + arx.z; a_rx.y = arx.y + arx.w;

    v2f ones; ones.x = 1.0f; ones.y = 1.0f;  // B = 4x16 all-ones
    v8f cz = {};

    // D[m,n] = sum_k A[m,k]  (replicated across n) : per-row reduction.
    v8f d_rr = __builtin_amdgcn_wmma_f32_16x16x4_f32(
        false, a_rr, false, ones, (short)0, cz, false, false);
    v8f d_xx = __builtin_amdgcn_wmma_f32_16x16x4_f32(
        false, a_xx, false, ones, (short)0, cz, false, false);
    v8f d_rx = __builtin_amdgcn_wmma_f32_16x16x4_f32(
        false, a_rx, false, ones, (short)0, cz, false, false);

    // C/D layout: VGPR v, lanes 0-15 -> M=v ; lanes 16-31 -> M=v+8.
    // Values are identical across N, so lane 0 carries rows 0..7 and
    // lane 16 carries rows 8..15.
    if ((lane & 15) == 0) {
        int base = wave * 16 + (lane >> 4) * 8;
#pragma unroll
        for (int v = 0; v < 8; ++v) {
            s_rr[base + v] = d_rr[v];
            s_xx[base + v] = d_xx[v];
            s_rx[base + v] = d_rx[v];
        }
    }
    __syncthreads();

    if (tid < 16) {
        float rr = 0.f, xx = 0.f, rx = 0.f;
#pragma unroll
        for (int w = 0; w < WAVES_PER_BLOCK; ++w) {
            rr += s_rr[w * 16 + tid];
            xx += s_xx[w * 16 + tid];
            rx += s_rx[w * 16 + tid];
        }
        float cosv = rx / sqrtf(rr * xx);
        cosv = fminf(1.0f, fmaxf(-1.0f, cosv));
        s_theta[tid] = acosf(cosv);
    }
    __syncthreads();

    if (tid == 0) {
        float s = 0.f;
#pragma unroll
        for (int r = 0; r < 16; ++r) s += s_theta[r];
        blockPartial[blockIdx.x] = s;
    }
}

// Kernel 2: deterministic single-block tree reduction of block partials.
__global__ __launch_bounds__(256) void pal_reduce_kernel(
    const float* __restrict__ part, int nPart, float* __restrict__ out,
    float invN)
{
    __shared__ float s[256];
    float v = 0.f;
    for (int i = threadIdx.x; i < nPart; i += 256) v += part[i];
    s[threadIdx.x] = v;
    __syncthreads();
#pragma unroll
    for (int off = 128; off > 0; off >>= 1) {
        if ((int)threadIdx.x < off) s[threadIdx.x] += s[threadIdx.x + off];
        __syncthreads();
    }
    if (threadIdx.x == 0) out[0] = s[0] * invN;
}

extern "C" void kernel_launch(void* const* d_in, const int* in_sizes, int n_in,
                              void* d_out, int out_size, void* d_ws, size_t ws_size,
                              hipStream_t stream) {
    const float* recon = (const float*)d_in[0];
    const float* x     = (const float*)d_in[1];
    const int total = in_sizes[0];
    const int N = total / D_DIM;               // 16384
    const int nBlocks = N / ROWS_PER_BLOCK;    // 1024

    float* part = (float*)d_ws;                // 4 KB of workspace

    pal_rowdot_kernel<<<nBlocks, 256, 0, stream>>>(recon, x, part);
    pal_reduce_kernel<<<1, 256, 0, stream>>>(part, nBlocks, (float*)d_out,
                                             1.0f / (float)N);
}